// NAGphormer_25228637897374
// MI455X (gfx1250) — compile-verified
//
#include <hip/hip_runtime.h>
#include <math.h>

typedef float v2f __attribute__((ext_vector_type(2)));
typedef float v8f __attribute__((ext_vector_type(8)));

#define HID 64
#define OUTC 40

// ---------------------------------------------------------------------------
// Degree / normalization kernels
// ---------------------------------------------------------------------------
__global__ void init_deg_kernel(float* deg, int N) {
    int i = blockIdx.x * blockDim.x + threadIdx.x;
    if (i < N) deg[i] = 1.0f;  // self-loop contribution
}

__global__ void edge_deg_kernel(const int* __restrict__ dst, float* deg, int E) {
    int e = blockIdx.x * blockDim.x + threadIdx.x;
    if (e < E) atomicAdd(&deg[dst[e]], 1.0f);
}

__global__ void dinv_kernel(const float* __restrict__ deg, float* dinv, int N) {
    int i = blockIdx.x * blockDim.x + threadIdx.x;
    if (i < N) {
        float d = deg[i];
        dinv[i] = (d > 0.0f) ? rsqrtf(d) : 0.0f;
    }
}

// pnext = p * dinv^2  (self loop term; also serves as full initialization)
__global__ void self_prop_kernel(const float* __restrict__ p,
                                 const float* __restrict__ dinv,
                                 float* __restrict__ pnext, int N) {
    long long idx = (long long)blockIdx.x * blockDim.x + threadIdx.x;
    if (idx < (long long)N * HID) {
        int i = (int)(idx >> 6);
        float d = dinv[i];
        pnext[idx] = p[idx] * d * d;
    }
}

// pnext[dst] += p[src] * dinv[src]*dinv[dst], one thread per (edge, feature)
__global__ void edge_prop_kernel(const int* __restrict__ src,
                                 const int* __restrict__ dst,
                                 const float* __restrict__ dinv,
                                 const float* __restrict__ p,
                                 float* __restrict__ pnext, int E) {
    long long idx = (long long)blockIdx.x * blockDim.x + threadIdx.x;
    if (idx < (long long)E * HID) {
        int e = (int)(idx >> 6);
        int h = (int)(idx & 63);
        int s = __ldg(&src[e]);
        int d = __ldg(&dst[e]);
        float w = __ldg(&dinv[s]) * __ldg(&dinv[d]);
        atomicAdd(&pnext[(long long)d * HID + h], p[(long long)s * HID + h] * w);
    }
}

// ---------------------------------------------------------------------------
// WMMA fp32 GEMM specialized for Nout == 64:
//   out[M,64] = relu(A[M,K] @ W[64,K]^T + bias)
// Single K pass (A read once from HBM), four independent accumulator chains,
// fragments software-pipelined one k-step ahead so loads overlap the WMMAs.
// Block = 128 threads (4 waves), each wave owns one 16-row tile.
// ---------------------------------------------------------------------------
__global__ __launch_bounds__(128) void gemm_wmma_n64_relu(
    const float* __restrict__ A, const float* __restrict__ W,
    const float* __restrict__ bias, float* __restrict__ out,
    int M, int K) {
    int wave = threadIdx.x >> 5;
    int lane = threadIdx.x & 31;
    int hl = lane >> 4;       // lane half (0 or 1)
    int lr = lane & 15;       // lane within half
    int m0 = (blockIdx.x * 4 + wave) * 16;
    if (m0 + 16 > M) return;  // wave-uniform: EXEC stays all-ones for WMMA

    v8f c0 = {}, c1 = {}, c2 = {}, c3 = {};
    const float* ap  = A + (long long)(m0 + lr) * K + 2 * hl;
    const float* bp0 = W + (0 + lr) * K + 2 * hl;
    const float* bp1 = W + (16 + lr) * K + 2 * hl;
    const float* bp2 = W + (32 + lr) * K + 2 * hl;
    const float* bp3 = W + (48 + lr) * K + 2 * hl;

    // prologue: fragments for k = 0
    v2f a, b0, b1, b2, b3;
    a[0]  = ap[0];  a[1]  = ap[1];
    b0[0] = bp0[0]; b0[1] = bp0[1];
    b1[0] = bp1[0]; b1[1] = bp1[1];
    b2[0] = bp2[0]; b2[1] = bp2[1];
    b3[0] = bp3[0]; b3[1] = bp3[1];

    for (int k0 = 0; k0 < K - 4; k0 += 4) {
        // prefetch fragments for k0 + 4 while the WMMAs below execute
        v2f an, b0n, b1n, b2n, b3n;
        int kn = k0 + 4;
        an[0]  = ap[kn];  an[1]  = ap[kn + 1];
        b0n[0] = bp0[kn]; b0n[1] = bp0[kn + 1];
        b1n[0] = bp1[kn]; b1n[1] = bp1[kn + 1];
        b2n[0] = bp2[kn]; b2n[1] = bp2[kn + 1];
        b3n[0] = bp3[kn]; b3n[1] = bp3[kn + 1];

        c0 = __builtin_amdgcn_wmma_f32_16x16x4_f32(false, a, false, b0, (short)0, c0, false, false);
        c1 = __builtin_amdgcn_wmma_f32_16x16x4_f32(false, a, false, b1, (short)0, c1, false, false);
        c2 = __builtin_amdgcn_wmma_f32_16x16x4_f32(false, a, false, b2, (short)0, c2, false, false);
        c3 = __builtin_amdgcn_wmma_f32_16x16x4_f32(false, a, false, b3, (short)0, c3, false, false);

        a = an; b0 = b0n; b1 = b1n; b2 = b2n; b3 = b3n;
    }
    // epilogue: last k-step
    c0 = __builtin_amdgcn_wmma_f32_16x16x4_f32(false, a, false, b0, (short)0, c0, false, false);
    c1 = __builtin_amdgcn_wmma_f32_16x16x4_f32(false, a, false, b1, (short)0, c1, false, false);
    c2 = __builtin_amdgcn_wmma_f32_16x16x4_f32(false, a, false, b2, (short)0, c2, false, false);
    c3 = __builtin_amdgcn_wmma_f32_16x16x4_f32(false, a, false, b3, (short)0, c3, false, false);

    float bv0 = bias[0 + lr], bv1 = bias[16 + lr];
    float bv2 = bias[32 + lr], bv3 = bias[48 + lr];
#pragma unroll
    for (int i = 0; i < 8; ++i) {
        long long row = (long long)(m0 + i + 8 * hl) * HID;
        out[row + 0 + lr]  = fmaxf(c0[i] + bv0, 0.0f);
        out[row + 16 + lr] = fmaxf(c1[i] + bv1, 0.0f);
        out[row + 32 + lr] = fmaxf(c2[i] + bv2, 0.0f);
        out[row + 48 + lr] = fmaxf(c3[i] + bv3, 0.0f);
    }
}

// ---------------------------------------------------------------------------
// WMMA helper operating on LDS tiles: sOut = act(sA[64,K] @ W[Nout,K]^T + b)
// Called by all 128 threads; wave w handles rows [w*16, w*16+16).
// act: 0 none, 1 relu, 2 gelu(exact erf)
// ---------------------------------------------------------------------------
__device__ __forceinline__ void lds_gemm(const float* sA, int lda,
                                         const float* __restrict__ W,
                                         const float* __restrict__ bias,
                                         float* sOut, int ldo,
                                         int K, int Nout, int act) {
    int wave = threadIdx.x >> 5;
    int lane = threadIdx.x & 31;
    int hl = lane >> 4;
    int lr = lane & 15;
    int r0 = wave * 16;

    for (int n0 = 0; n0 < Nout; n0 += 16) {
        v8f c = {};
        for (int k0 = 0; k0 < K; k0 += 4) {
            v2f a, b;
            a[0] = sA[(r0 + lr) * lda + k0 + 2 * hl];
            a[1] = sA[(r0 + lr) * lda + k0 + 1 + 2 * hl];
            const float* bp = W + (n0 + lr) * K + k0 + 2 * hl;
            b[0] = bp[0]; b[1] = bp[1];
            c = __builtin_amdgcn_wmma_f32_16x16x4_f32(false, a, false, b,
                                                      (short)0, c, false, false);
        }
        float bv = bias[n0 + lr];
#pragma unroll
        for (int i = 0; i < 8; ++i) {
            float v = c[i] + bv;
            if (act == 1) v = fmaxf(v, 0.0f);
            else if (act == 2) v = 0.5f * v * (1.0f + erff(v * 0.70710678118654752f));
            sOut[(r0 + i + 8 * hl) * ldo + n0 + lr] = v;
        }
    }
}

// ---------------------------------------------------------------------------
// Fused transformer layer + mean-pool + classifier.
// 16 nodes per block (64 seq rows), 128 threads, dynamic LDS.
// ---------------------------------------------------------------------------
#define LD_SEQ 68
#define LD_QKV 200
#define LD_TMP 132

__global__ __launch_bounds__(128) void transformer_kernel(
    const float* __restrict__ tok0, const float* __restrict__ tok1,
    const float* __restrict__ tok2, const float* __restrict__ tok3,
    const float* __restrict__ in_proj_W, const float* __restrict__ in_proj_b,
    const float* __restrict__ out_W, const float* __restrict__ out_b,
    const float* __restrict__ ff_W1, const float* __restrict__ ff_b1,
    const float* __restrict__ ff_W2, const float* __restrict__ ff_b2,
    const float* __restrict__ ln1_w, const float* __restrict__ ln1_b,
    const float* __restrict__ ln2_w, const float* __restrict__ ln2_b,
    const float* __restrict__ cls_W, const float* __restrict__ cls_b,
    float* __restrict__ out, int N) {
    extern __shared__ float smem[];
    float* sSeq = smem;                       // [64][LD_SEQ]
    float* sQkv = sSeq + 64 * LD_SEQ;         // [64][LD_QKV]
    float* sTmp = sQkv + 64 * LD_QKV;         // [64][LD_TMP]

    int node0 = blockIdx.x * 16;
    int tid = threadIdx.x;

    // ---- load 64 rows x 64 cols of the token sequence into LDS ----
    for (int i = tid; i < 64 * 64; i += 128) {
        int r = i >> 6, h = i & 63;
        int node = node0 + (r >> 2), t = r & 3;
        const float* tp = (t == 0) ? tok0 : (t == 1) ? tok1 : (t == 2) ? tok2 : tok3;
        sSeq[r * LD_SEQ + h] = tp[(long long)node * HID + h];
    }
    __syncthreads();

    // ---- qkv projection: [64,64] @ [64,192]^T -> sQkv ----
    lds_gemm(sSeq, LD_SEQ, in_proj_W, in_proj_b, sQkv, LD_QKV, HID, 3 * HID, 0);
    __syncthreads();

    // ---- attention (seq=4, nhead=1): thread r in [0,64) owns one (node,s) row
    if (tid < 64) {
        int rbase = (tid >> 2) * 4;
        int s = tid & 3;
        float sc[4];
        float mx = -1e30f;
        for (int t = 0; t < 4; ++t) {
            float acc = 0.0f;
            for (int h = 0; h < HID; ++h)
                acc += sQkv[(rbase + s) * LD_QKV + h] *
                       sQkv[(rbase + t) * LD_QKV + HID + h];
            sc[t] = acc * 0.125f;  // 1/sqrt(64)
            mx = fmaxf(mx, sc[t]);
        }
        float se = 0.0f;
        for (int t = 0; t < 4; ++t) { sc[t] = __expf(sc[t] - mx); se += sc[t]; }
        float rs = 1.0f / se;
        for (int h = 0; h < HID; ++h) {
            float a = 0.0f;
            for (int t = 0; t < 4; ++t)
                a += sc[t] * rs * sQkv[(rbase + t) * LD_QKV + 2 * HID + h];
            sTmp[(rbase + s) * LD_TMP + h] = a;
        }
    }
    __syncthreads();

    // ---- output projection -> sQkv[:, 0:64] ----
    lds_gemm(sTmp, LD_TMP, out_W, out_b, sQkv, LD_QKV, HID, HID, 0);
    __syncthreads();

    // ---- residual + LayerNorm1 -> sSeq ----
    if (tid < 64) {
        int r = tid;
        float m = 0.0f;
        for (int h = 0; h < HID; ++h)
            m += sSeq[r * LD_SEQ + h] + sQkv[r * LD_QKV + h];
        m *= (1.0f / HID);
        float v = 0.0f;
        for (int h = 0; h < HID; ++h) {
            float d = sSeq[r * LD_SEQ + h] + sQkv[r * LD_QKV + h] - m;
            v += d * d;
        }
        float rinv = rsqrtf(v * (1.0f / HID) + 1e-5f);
        for (int h = 0; h < HID; ++h) {
            float d = sSeq[r * LD_SEQ + h] + sQkv[r * LD_QKV + h] - m;
            sSeq[r * LD_SEQ + h] = d * rinv * ln1_w[h] + ln1_b[h];
        }
    }
    __syncthreads();

    // ---- FFN: gelu(seq @ ff_W1^T + b1) -> sTmp[64,128] ----
    lds_gemm(sSeq, LD_SEQ, ff_W1, ff_b1, sTmp, LD_TMP, HID, 2 * HID, 2);
    __syncthreads();

    // ---- FFN second half -> sQkv[:, 0:64] ----
    lds_gemm(sTmp, LD_TMP, ff_W2, ff_b2, sQkv, LD_QKV, 2 * HID, HID, 0);
    __syncthreads();

    // ---- residual + LayerNorm2 -> sSeq ----
    if (tid < 64) {
        int r = tid;
        float m = 0.0f;
        for (int h = 0; h < HID; ++h)
            m += sSeq[r * LD_SEQ + h] + sQkv[r * LD_QKV + h];
        m *= (1.0f / HID);
        float v = 0.0f;
        for (int h = 0; h < HID; ++h) {
            float d = sSeq[r * LD_SEQ + h] + sQkv[r * LD_QKV + h] - m;
            v += d * d;
        }
        float rinv = rsqrtf(v * (1.0f / HID) + 1e-5f);
        for (int h = 0; h < HID; ++h) {
            float d = sSeq[r * LD_SEQ + h] + sQkv[r * LD_QKV + h] - m;
            sSeq[r * LD_SEQ + h] = d * rinv * ln2_w[h] + ln2_b[h];
        }
    }
    __syncthreads();

    // ---- mean over tokens -> sTmp[0 .. 16*64) ----
    for (int i = tid; i < 16 * HID; i += 128) {
        int nl = i >> 6, h = i & 63;
        int rb = nl * 4;
        sTmp[i] = 0.25f * (sSeq[(rb + 0) * LD_SEQ + h] + sSeq[(rb + 1) * LD_SEQ + h] +
                           sSeq[(rb + 2) * LD_SEQ + h] + sSeq[(rb + 3) * LD_SEQ + h]);
    }
    __syncthreads();

    // ---- classifier: 16 nodes x 40 outputs ----
    for (int i = tid; i < 16 * OUTC; i += 128) {
        int nl = i / OUTC, o = i % OUTC;
        float acc = cls_b[o];
        for (int h = 0; h < HID; ++h)
            acc += sTmp[nl * HID + h] * cls_W[o * HID + h];
        out[(long long)(node0 + nl) * OUTC + o] = acc;
    }
}

// ---------------------------------------------------------------------------
// Host-side orchestration
// ---------------------------------------------------------------------------
extern "C" void kernel_launch(void* const* d_in, const int* in_sizes, int n_in,
                              void* d_out, int out_size, void* d_ws, size_t ws_size,
                              hipStream_t stream) {
    const int IN = 256;
    const int N = in_sizes[0] / IN;       // 100000
    const int E = in_sizes[1] / 2;        // 1600000

    const float* x        = (const float*)d_in[0];
    const int*   ei       = (const int*)d_in[1];
    const int*   src      = ei;
    const int*   dst      = ei + E;
    const float* W_in     = (const float*)d_in[2];
    const float* b_in     = (const float*)d_in[3];
    const float* sg_W     = (const float*)d_in[4];
    const float* sg_b     = (const float*)d_in[5];
    const float* in_proj_W = (const float*)d_in[6];
    const float* in_proj_b = (const float*)d_in[7];
    const float* out_W    = (const float*)d_in[8];
    const float* out_b    = (const float*)d_in[9];
    const float* ff_W1    = (const float*)d_in[10];
    const float* ff_b1    = (const float*)d_in[11];
    const float* ff_W2    = (const float*)d_in[12];
    const float* ff_b2    = (const float*)d_in[13];
    const float* ln1_w    = (const float*)d_in[14];
    const float* ln1_b    = (const float*)d_in[15];
    const float* ln2_w    = (const float*)d_in[16];
    const float* ln2_b    = (const float*)d_in[17];
    const float* cls_W    = (const float*)d_in[18];
    const float* cls_b    = (const float*)d_in[19];
    float* out = (float*)d_out;

    // Workspace layout (floats)
    float* ws   = (float*)d_ws;
    long long NH = (long long)N * HID;
    float* deg  = ws;
    float* dinv = deg + N;
    float* tok0 = dinv + N;
    float* tok1 = tok0 + NH;
    float* tok2 = tok1 + NH;
    float* tok3 = tok2 + NH;
    float* pA   = tok3 + NH;
    float* pB   = pA + NH;

    const int T = 256;
    int gN   = (N + T - 1) / T;
    int gE   = (E + T - 1) / T;
    long long nElemsN = NH;
    long long nElemsE = (long long)E * HID;
    int gNH  = (int)((nElemsN + T - 1) / T);
    int gEH  = (int)((nElemsE + T - 1) / T);
    int gGemm = (N / 16 + 3) / 4;   // 4 row tiles per block
    int gTrans = N / 16;

    // degrees + normalization
    init_deg_kernel<<<gN, T, 0, stream>>>(deg, N);
    edge_deg_kernel<<<gE, T, 0, stream>>>(dst, deg, E);
    dinv_kernel<<<gN, T, 0, stream>>>(deg, dinv, N);

    // h0 = relu(x @ W_in^T + b_in)
    gemm_wmma_n64_relu<<<gGemm, 128, 0, stream>>>(x, W_in, b_in, tok0, N, IN);

    // hop 1: tok0 -> pA
    self_prop_kernel<<<gNH, T, 0, stream>>>(tok0, dinv, pA, N);
    edge_prop_kernel<<<gEH, T, 0, stream>>>(src, dst, dinv, tok0, pA, E);
    gemm_wmma_n64_relu<<<gGemm, 128, 0, stream>>>(pA, sg_W, sg_b, tok1, N, HID);

    // hop 2: pA -> pB
    self_prop_kernel<<<gNH, T, 0, stream>>>(pA, dinv, pB, N);
    edge_prop_kernel<<<gEH, T, 0, stream>>>(src, dst, dinv, pA, pB, E);
    gemm_wmma_n64_relu<<<gGemm, 128, 0, stream>>>(pB, sg_W + HID * HID, sg_b + HID,
                                                  tok2, N, HID);

    // hop 3: pB -> pA
    self_prop_kernel<<<gNH, T, 0, stream>>>(pB, dinv, pA, N);
    edge_prop_kernel<<<gEH, T, 0, stream>>>(src, dst, dinv, pB, pA, E);
    gemm_wmma_n64_relu<<<gGemm, 128, 0, stream>>>(pA, sg_W + 2 * HID * HID,
                                                  sg_b + 2 * HID, tok3, N, HID);

    // fused transformer + classifier
    size_t smemBytes = (size_t)(64 * LD_SEQ + 64 * LD_QKV + 64 * LD_TMP) * sizeof(float);
    transformer_kernel<<<gTrans, 128, smemBytes, stream>>>(
        tok0, tok1, tok2, tok3, in_proj_W, in_proj_b, out_W, out_b,
        ff_W1, ff_b1, ff_W2, ff_b2, ln1_w, ln1_b, ln2_w, ln2_b,
        cls_W, cls_b, out, N);
}